// TimeAttention_33715493274067
// MI455X (gfx1250) — compile-verified
//
#include <hip/hip_runtime.h>
#include <cstdint>

// Problem constants (match reference)
#define BB    32      // batch
#define TT    512     // targets per batch
#define CCN   512     // context per batch
#define HALF  50
#define WW    101     // window size
#define WPAD  104     // padded curve stride in LDS (keeps 16B alignment)

#define ROWS    8               // waves (rows) per block
#define THREADS (ROWS * 32)     // 256 threads = 8 wave32

// ---- CDNA5 async global->LDS copy (ASYNCcnt-tracked, no VGPR round trip) ----
// VGLOBAL encoding, GV mode: per-lane LDS byte address in a VGPR, 64-bit
// global address in a VGPR pair. Tracked by ASYNCcnt.
__device__ __forceinline__ void async_ld_b32(uint32_t lds_byte_off, const void* gaddr) {
  asm volatile("global_load_async_to_lds_b32 %0, %1, off"
               :: "v"(lds_byte_off), "v"(gaddr)
               : "memory");
}
__device__ __forceinline__ void wait_async0() {
  asm volatile("s_wait_asynccnt 0" ::: "memory");
}
// Low 32 bits of a flat pointer into LDS are the LDS byte address
// (ISA aperture rule).
__device__ __forceinline__ uint32_t lds_off(const void* p) {
  return (uint32_t)(uintptr_t)p;
}

__global__ __launch_bounds__(THREADS) void time_attn_softmax_kernel(
    const int*   __restrict__ concepts,  // [B,T]  (harness passes integers as i32)
    const int*   __restrict__ tgt_ts,    // [B,T]
    const int*   __restrict__ ctx_ts,    // [B,C]
    const int*   __restrict__ time_mask, // [B,C]  0/1
    const float* __restrict__ emb,       // [V,W]
    const float* __restrict__ bias,      // [W]
    float*       __restrict__ out)       // [B,T,C]
{
  __shared__ int   s_ctx[CCN];            // context timestamps for this b
  __shared__ int   s_msk[CCN];            // time mask for this b
  __shared__ float s_curve[ROWS][WPAD];   // per-wave time-attention curve

  const int tid  = threadIdx.x;
  const int lane = tid & 31;
  const int wv   = tid >> 5;

  const int blocks_per_b = TT / ROWS;                    // 64
  const int b = blockIdx.x / blocks_per_b;
  const int t = (blockIdx.x % blocks_per_b) * ROWS + wv; // this wave's target row
  const int rowidx = b * TT + t;

  // ---- Stage ctx timestamps + mask for batch b via async DMA to LDS ----
  {
    const int* crow = ctx_ts    + (size_t)b * CCN;
    const int* mrow = time_mask + (size_t)b * CCN;
    const uint32_t coff = lds_off(&s_ctx[0]);
    const uint32_t moff = lds_off(&s_msk[0]);
#pragma unroll
    for (int j = 0; j < CCN / THREADS; ++j) {            // 2 iters
      const int c = tid + j * THREADS;
      async_ld_b32(coff + 4u * (uint32_t)c, crow + c);
      async_ld_b32(moff + 4u * (uint32_t)c, mrow + c);
    }
  }

  // ---- Stage this row's embedding curve (101 f32) via async DMA ----
  const int concept = concepts[rowidx];
  const float* erow = emb + (size_t)concept * WW;
  {
    const uint32_t koff = lds_off(&s_curve[wv][0]);
#pragma unroll
    for (int j = 0; j < 4; ++j) {
      const int i = lane + 32 * j;
      if (i < WW) async_ld_b32(koff + 4u * (uint32_t)i, erow + i);
    }
  }

  wait_async0();        // this wave's async copies are in LDS
  __syncthreads();      // everyone's copies visible to the whole block

  // ---- Fold bias into the curve (per-wave region) ----
#pragma unroll
  for (int j = 0; j < 4; ++j) {
    const int i = lane + 32 * j;
    if (i < WW) s_curve[wv][i] += bias[i];
  }
  __syncthreads();

  // ---- Gather logits + online row softmax (C=512 -> 16 cols per lane) ----
  const int tts = tgt_ts[rowidx];
  float l[16];
  float m = -3.0e38f;
#pragma unroll
  for (int j = 0; j < 16; ++j) {
    const int c = lane + 32 * j;            // conflict-free bank pattern for s_ctx/s_msk
    int d = s_ctx[c] - tts;
    d = (d < -HALF) ? -HALF : ((d > HALF) ? HALF : d);
    float v = s_curve[wv][d + HALF];        // data-dependent LDS gather
    v -= 1.0e9f * (float)s_msk[c];
    l[j] = v;
    m = fmaxf(m, v);
  }
  // wave32 max-reduce
#pragma unroll
  for (int off = 16; off > 0; off >>= 1)
    m = fmaxf(m, __shfl_xor(m, off, 32));

  float s = 0.0f;
#pragma unroll
  for (int j = 0; j < 16; ++j) {
    const float e = __expf(l[j] - m);
    l[j] = e;
    s += e;
  }
  // wave32 sum-reduce
#pragma unroll
  for (int off = 16; off > 0; off >>= 1)
    s += __shfl_xor(s, off, 32);
  const float inv = __builtin_amdgcn_rcpf(s);

  // ---- Coalesced store of the softmax row ----
  float* orow = out + (size_t)rowidx * CCN;
#pragma unroll
  for (int j = 0; j < 16; ++j)
    orow[lane + 32 * j] = l[j] * inv;
}

extern "C" void kernel_launch(void* const* d_in, const int* in_sizes, int n_in,
                              void* d_out, int out_size, void* d_ws, size_t ws_size,
                              hipStream_t stream) {
  (void)in_sizes; (void)n_in; (void)out_size; (void)d_ws; (void)ws_size;
  const int*   concepts  = (const int*)d_in[0];   // target_concepts
  const int*   tgt_ts    = (const int*)d_in[1];   // target_time_stamps
  const int*   ctx_ts    = (const int*)d_in[2];   // context_time_stamps
  const int*   time_mask = (const int*)d_in[3];   // time_mask
  const float* emb       = (const float*)d_in[4]; // embedding [V,W]
  const float* bias      = (const float*)d_in[5]; // bias [W]
  float*       out       = (float*)d_out;         // [B,T,C]

  dim3 grid(BB * (TT / ROWS));   // 2048 blocks
  dim3 block(THREADS);           // 256 threads = 8 wave32
  time_attn_softmax_kernel<<<grid, block, 0, stream>>>(
      concepts, tgt_ts, ctx_ts, time_mask, emb, bias, out);
}